// GCN_1219770712798
// MI455X (gfx1250) — compile-verified
//
#include <hip/hip_runtime.h>

typedef __attribute__((ext_vector_type(2))) float v2f;
typedef __attribute__((ext_vector_type(8))) float v8f;

#define FD 128   // feature dim

// Guaranteed single-instruction fp32 atomic add (no return): resolves at L2.
__device__ __forceinline__ void atomAddF(float* p, float v) {
  asm volatile("global_atomic_add_f32 %0, %1, off" :: "v"(p), "v"(v) : "memory");
}

// ---------------------------------------------------------------------------
// Dual GEMM: Mout = X@W ; Rout = relu(X@RW + rb). AFFINE folds the previous
// layer's BatchNorm into the A-operand load (compile-time, branch-free).
// Wave tile: 16 rows x 16 cols, both weight matrices share the A operand.
// Block: 32x8 (8 waves) -> 16 rows x 128 cols per block. Grid: N/16.
// Uses V_WMMA_F32_16X16X4_F32 (full fp32, matches reference precision).
// ---------------------------------------------------------------------------
template <bool AFFINE>
__global__ void __launch_bounds__(256)
gemm_dual_wmma(const float* __restrict__ X, const float* __restrict__ W,
               const float* __restrict__ RW, const float* __restrict__ rbias,
               const float* __restrict__ scale, const float* __restrict__ shift,
               float* __restrict__ Mout, float* __restrict__ Rout) {
  const int lane  = threadIdx.x;        // 0..31
  const int wave  = threadIdx.y;        // 0..7 -> 16-col tile
  const int row0  = blockIdx.x * 16;
  const int col0  = wave * 16;
  const int mrow  = lane & 15;          // A: M index
  const int khalf = (lane >> 4) << 1;   // A/B: 0 or 2 (K sub-block per half-wave)
  const int n     = lane & 15;          // B/C/D: N index

  const float* xrow = X + (size_t)(row0 + mrow) * FD;

  v8f accM = {};
  v8f accR = {};

#pragma unroll 8
  for (int k = 0; k < FD; k += 4) {
    const int ka = k + khalf;
    float a0 = xrow[ka];
    float a1 = xrow[ka + 1];
    if (AFFINE) {
      a0 = a0 * scale[ka]     + shift[ka];
      a1 = a1 * scale[ka + 1] + shift[ka + 1];
    }
    v2f a; a.x = a0; a.y = a1;

    v2f bw, br;
    bw.x = W [(size_t)ka       * FD + col0 + n];
    bw.y = W [(size_t)(ka + 1) * FD + col0 + n];
    br.x = RW[(size_t)ka       * FD + col0 + n];
    br.y = RW[(size_t)(ka + 1) * FD + col0 + n];

    accM = __builtin_amdgcn_wmma_f32_16x16x4_f32(false, a, false, bw,
                                                 (short)0, accM, false, false);
    accR = __builtin_amdgcn_wmma_f32_16x16x4_f32(false, a, false, br,
                                                 (short)0, accR, false, false);
  }

  const int rowAdd = (lane >> 4) << 3;  // lanes 16-31 hold M+8
#pragma unroll
  for (int i = 0; i < 8; ++i) {
    const int row = row0 + rowAdd + i;
    const size_t idx = (size_t)row * FD + col0 + n;
    Mout[idx] = accM[i];
    float rv = accR[i] + rbias[col0 + n];
    Rout[idx] = rv > 0.0f ? rv : 0.0f;
  }
}

// ---------------------------------------------------------------------------
// Edge scatter: agg[dst[e], :] += m[src[e], :]. One thread = (edge, 4 cols).
// m and agg both fit in the 192MB L2 -> atomics resolve at L2.
// ---------------------------------------------------------------------------
__global__ void __launch_bounds__(256)
edge_scatter(const float* __restrict__ M, const int* __restrict__ src,
             const int* __restrict__ dst, float* __restrict__ AGG, int total) {
  const int idx = blockIdx.x * 256 + threadIdx.x;
  if (idx >= total) return;
  const int e = idx >> 5;           // edge id (32 threads per edge)
  const int c = (idx & 31) << 2;    // 4 columns per thread
  const int s = src[e];
  const int d = dst[e];
  const float4 v = *(const float4*)(M + (size_t)s * FD + c);
  float* p = AGG + (size_t)d * FD + c;
  atomAddF(p + 0, v.x);
  atomAddF(p + 1, v.y);
  atomAddF(p + 2, v.z);
  atomAddF(p + 3, v.w);
}

// ---------------------------------------------------------------------------
// Combine: h = relu(agg + b) + r   (in-place into AGG)
// plus per-column sum / sum-of-squares partials for BatchNorm.
// Block handles 32 rows x 128 cols; 2 threads per column.
// ---------------------------------------------------------------------------
__global__ void __launch_bounds__(256)
combine_stats(float* __restrict__ AGG, const float* __restrict__ bias,
              const float* __restrict__ Rr, float* __restrict__ statSum,
              float* __restrict__ statSq, int Nrows) {
  const int col = threadIdx.x & 127;
  const int rh  = threadIdx.x >> 7;    // 0 or 1
  const int base = blockIdx.x * 32;
  float s = 0.0f, q = 0.0f;
  const float b = bias[col];
#pragma unroll 4
  for (int rr = rh; rr < 32; rr += 2) {
    const int row = base + rr;
    if (row >= Nrows) break;
    const size_t idx = (size_t)row * FD + col;
    float v = AGG[idx] + b;
    v = v > 0.0f ? v : 0.0f;
    v += Rr[idx];
    AGG[idx] = v;
    s += v;
    q += v * v;
  }
  __shared__ float ls[128];
  __shared__ float lq[128];
  if (rh == 1) { ls[col] = s; lq[col] = q; }
  __syncthreads();
  if (rh == 0) {
    atomAddF(&statSum[col], s + ls[col]);
    atomAddF(&statSq[col],  q + lq[col]);
  }
}

// ---------------------------------------------------------------------------
// Finalize BN stats: scale = gamma * rsqrt(var+eps); shift = beta - mu*scale.
// ---------------------------------------------------------------------------
__global__ void bn_finalize(const float* __restrict__ statSum,
                            const float* __restrict__ statSq,
                            const float* __restrict__ g,
                            const float* __restrict__ beta,
                            float* __restrict__ scale,
                            float* __restrict__ shift, float invN) {
  const int c = threadIdx.x;  // 128 threads
  const float mu  = statSum[c] * invN;
  const float var = statSq[c] * invN - mu * mu;
  const float rs  = rsqrtf(var + 1e-5f);
  const float sc  = rs * g[c];
  scale[c] = sc;
  shift[c] = beta[c] - mu * sc;
}

// ---------------------------------------------------------------------------
// Final normalize: out = h*scale + shift (vectorized float4).
// ---------------------------------------------------------------------------
__global__ void __launch_bounds__(256)
bn_apply(const float* __restrict__ H, const float* __restrict__ scale,
         const float* __restrict__ shift, float* __restrict__ out, int nd4) {
  const int i = blockIdx.x * 256 + threadIdx.x;
  if (i >= nd4) return;
  float4 v = ((const float4*)H)[i];
  const int c = (i << 2) & 127;
  v.x = v.x * scale[c + 0] + shift[c + 0];
  v.y = v.y * scale[c + 1] + shift[c + 1];
  v.z = v.z * scale[c + 2] + shift[c + 2];
  v.w = v.w * scale[c + 3] + shift[c + 3];
  ((float4*)out)[i] = v;
}

// ---------------------------------------------------------------------------
extern "C" void kernel_launch(void* const* d_in, const int* in_sizes, int n_in,
                              void* d_out, int out_size, void* d_ws, size_t ws_size,
                              hipStream_t stream) {
  const float* feats = (const float*)d_in[0];
  const int*   src   = (const int*)  d_in[1];
  const int*   dst   = (const int*)  d_in[2];
  const float* W0    = (const float*)d_in[3];
  const float* b0    = (const float*)d_in[4];
  const float* rW0   = (const float*)d_in[5];
  const float* rb0   = (const float*)d_in[6];
  const float* g0    = (const float*)d_in[7];
  const float* be0   = (const float*)d_in[8];
  const float* W1    = (const float*)d_in[9];
  const float* b1    = (const float*)d_in[10];
  const float* rW1   = (const float*)d_in[11];
  const float* rb1   = (const float*)d_in[12];
  const float* g1    = (const float*)d_in[13];
  const float* be1   = (const float*)d_in[14];

  const int N = in_sizes[0] / FD;    // 100000
  const int E = in_sizes[1];         // 1600000
  const size_t ND = (size_t)N * FD;

  float* m    = (float*)d_ws;                 // [N,128]
  float* r    = m + ND;                       // [N,128]
  float* agg  = r + ND;                       // [N,128]  (also holds h in-place)
  float* stat = agg + ND;                     // sum[128] sq[128] scale[128] shift[128]
  float* statSum = stat;
  float* statSq  = stat + 128;
  float* scl     = stat + 256;
  float* shf     = stat + 384;

  const dim3 gemmBlk(32, 8);
  const int  gemmGrid   = N / 16;                       // 6250
  const int  scatTotal  = E * 32;
  const int  scatGrid   = (scatTotal + 255) / 256;      // 200000
  const int  combGrid   = (N + 31) / 32;                // 3125
  const int  nd4        = (int)(ND / 4);
  const int  applyGrid  = (nd4 + 255) / 256;
  const float invN = 1.0f / (float)N;

  // -------- Layer 1 --------
  hipMemsetAsync(agg, 0, ND * sizeof(float), stream);
  hipMemsetAsync(statSum, 0, 256 * sizeof(float), stream);
  gemm_dual_wmma<false><<<gemmGrid, gemmBlk, 0, stream>>>(feats, W0, rW0, rb0,
                                                          nullptr, nullptr, m, r);
  edge_scatter<<<scatGrid, 256, 0, stream>>>(m, src, dst, agg, scatTotal);
  combine_stats<<<combGrid, 256, 0, stream>>>(agg, b0, r, statSum, statSq, N);
  bn_finalize<<<1, 128, 0, stream>>>(statSum, statSq, g0, be0, scl, shf, invN);

  // -------- Layer 2 (BN of layer 1 folded into GEMM A-load) --------
  gemm_dual_wmma<true><<<gemmGrid, gemmBlk, 0, stream>>>(agg, W1, rW1, rb1,
                                                         scl, shf, m, r);
  hipMemsetAsync(agg, 0, ND * sizeof(float), stream);   // h0 consumed; reuse
  hipMemsetAsync(statSum, 0, 256 * sizeof(float), stream);
  edge_scatter<<<scatGrid, 256, 0, stream>>>(m, src, dst, agg, scatTotal);
  combine_stats<<<combGrid, 256, 0, stream>>>(agg, b1, r, statSum, statSq, N);
  bn_finalize<<<1, 128, 0, stream>>>(statSum, statSq, g1, be1, scl, shf, invN);
  bn_apply<<<applyGrid, 256, 0, stream>>>(agg, scl, shf, (float*)d_out, nd4);
}